// Attention_41566693491235
// MI455X (gfx1250) — compile-verified
//
#include <hip/hip_runtime.h>
#include <hip/hip_bf16.h>
#include <math.h>

#define B_ 256
#define L_ 196
#define R_ 1024
#define A_ 512
#define D_ 2048

typedef __attribute__((ext_vector_type(2))) float v2f;
typedef __attribute__((ext_vector_type(8))) float v8f;

// ---------------------------------------------------------------------------
// Kernel 1: att_h = h @ W_h + b_h   (256x1024 @ 1024x512 -> 256x512, fp32)
// One wave32 per 16x16 output tile, V_WMMA_F32_16X16X4_F32, K-loop of 256.
// A 16x4 layout: lanes 0-15 -> M=lane, VGPR{0,1}=K{0,1}; lanes 16-31 -> K{2,3}.
// B 4x16 layout: row K striped across lanes per VGPR (K=0/2 in v0, K=1/3 in v1).
// C/D: VGPR v -> M=v (lanes 0-15) / M=v+8 (lanes 16-31), N=lane&15.
// ---------------------------------------------------------------------------
__global__ __launch_bounds__(256) void h2att_wmma_kernel(
    const float* __restrict__ h, const float* __restrict__ W_h,
    const float* __restrict__ b_h, float* __restrict__ att_h)
{
    const int lane = threadIdx.x & 31;
    const int wave = threadIdx.x >> 5;
    const int tile = blockIdx.x * 8 + wave;        // 64 blocks * 8 waves = 512 tiles
    const int tM   = (tile >> 5) << 4;             // B rows: 16 tiles
    const int tN   = (tile & 31) << 4;             // A cols: 32 tiles
    const int half = lane >> 4;                    // 0: K 0..1, 1: K 2..3
    const int l16  = lane & 15;

    v8f acc = {};
    const float* arow = h + (size_t)(tM + l16) * R_;
    for (int k0 = 0; k0 < R_; k0 += 4) {
        const int ka = k0 + half * 2;
        v2f a; a.x = arow[ka]; a.y = arow[ka + 1];
        const float* bp = W_h + (size_t)ka * A_ + tN + l16;
        v2f b; b.x = bp[0]; b.y = bp[A_];
        acc = __builtin_amdgcn_wmma_f32_16x16x4_f32(
            /*neg_a=*/false, a, /*neg_b=*/false, b,
            /*c_mod=*/(short)0, acc, /*reuse_a=*/false, /*reuse_b=*/false);
    }
    const float bias = b_h[tN + l16];
#pragma unroll
    for (int v = 0; v < 8; ++v) {
        const int m = v + half * 8;
        att_h[(size_t)(tM + m) * A_ + tN + l16] = acc[v] + bias;
    }
}

// ---------------------------------------------------------------------------
// Kernel 2 (fused, one block per batch b):
//   dot[l]  = sum_a tanh(att_feats2[b,l,a] + att_h[b,a]) * w_a[a] + b_a
//   weight  = softmax(dot) * mask, renormalized   (softmax denom cancels)
//   out[b,d]= sum_l weight[l] * att_feats1[b,l,d]
// Streams att_feats2 (103MB) + att_feats1 (411MB) exactly once, float4 loads.
// ---------------------------------------------------------------------------
__global__ __launch_bounds__(256) void attn_fused_kernel(
    const float* __restrict__ att_feats1, const float* __restrict__ att_feats2,
    const float* __restrict__ att_masks,  const float* __restrict__ att_h,
    const float* __restrict__ w_a, const float* __restrict__ b_a_p,
    float* __restrict__ out)
{
    const int b    = blockIdx.x;
    const int t    = threadIdx.x;
    const int lane = t & 31;
    const int wave = t >> 5;

    __shared__ float4 s_atth[A_ / 4];   // 128 float4 = att_h[b,:]
    __shared__ float4 s_wa[A_ / 4];     // 128 float4 = w_a
    __shared__ float  s_dot[256];       // dot / weight (padded past L=196)
    __shared__ float  s_red[256];       // reduction scratch

    if (t < 128) {
        s_atth[t] = reinterpret_cast<const float4*>(att_h + (size_t)b * A_)[t];
    } else {
        s_wa[t - 128] = reinterpret_cast<const float4*>(w_a)[t - 128];
    }
    s_dot[t] = -INFINITY;
    __syncthreads();

    const float b_a = b_a_p[0];

    // ---- Phase A: per-row tanh-dot, one wave per row, lanes split A=512 ----
    for (int l = wave; l < L_; l += 8) {
        const float4* row =
            reinterpret_cast<const float4*>(att_feats2 + ((size_t)b * L_ + l) * A_);
        float s = 0.f;
#pragma unroll
        for (int it = 0; it < 4; ++it) {
            const int idx = it * 32 + lane;
            float4 x  = row[idx];
            float4 ah = s_atth[idx];
            float4 wa = s_wa[idx];
            s += tanhf(x.x + ah.x) * wa.x;
            s += tanhf(x.y + ah.y) * wa.y;
            s += tanhf(x.z + ah.z) * wa.z;
            s += tanhf(x.w + ah.w) * wa.w;
        }
#pragma unroll
        for (int off = 16; off > 0; off >>= 1) s += __shfl_xor(s, off, 32);
        if (lane == 0) s_dot[l] = s + b_a;
    }
    __syncthreads();

    // ---- Phase B: softmax + mask renormalization over L ----
    const float v = s_dot[t];            // -inf for t >= L_
    s_red[t] = v;
    __syncthreads();
    for (int off = 128; off > 0; off >>= 1) {
        if (t < off) s_red[t] = fmaxf(s_red[t], s_red[t + off]);
        __syncthreads();
    }
    const float vmax = s_red[0];
    __syncthreads();

    float e = 0.f;
    if (t < L_) e = expf(v - vmax) * att_masks[(size_t)b * L_ + t];
    s_red[t] = e;
    __syncthreads();
    for (int off = 128; off > 0; off >>= 1) {
        if (t < off) s_red[t] += s_red[t + off];
        __syncthreads();
    }
    const float denom = s_red[0];
    __syncthreads();
    s_dot[t] = e / denom;                // weight; 0 for t >= L_
    __syncthreads();

    // ---- Phase C: out[b, :] = sum_l weight[l] * att_feats1[b, l, :] ----
    float4 acc0 = {0.f, 0.f, 0.f, 0.f};
    float4 acc1 = {0.f, 0.f, 0.f, 0.f};
    const float4* base =
        reinterpret_cast<const float4*>(att_feats1 + (size_t)b * L_ * D_);
    for (int l = 0; l < L_; ++l) {
        const float  w   = s_dot[l];
        const float4* row = base + (size_t)l * (D_ / 4);
        __builtin_prefetch((const void*)(row + (D_ / 4)), 0, 0);
        float4 x0 = row[t];
        float4 x1 = row[t + 256];
        acc0.x += w * x0.x; acc0.y += w * x0.y; acc0.z += w * x0.z; acc0.w += w * x0.w;
        acc1.x += w * x1.x; acc1.y += w * x1.y; acc1.z += w * x1.z; acc1.w += w * x1.w;
    }
    float4* outp = reinterpret_cast<float4*>(out + (size_t)b * D_);
    outp[t]       = acc0;
    outp[t + 256] = acc1;
}

// ---------------------------------------------------------------------------
extern "C" void kernel_launch(void* const* d_in, const int* in_sizes, int n_in,
                              void* d_out, int out_size, void* d_ws, size_t ws_size,
                              hipStream_t stream)
{
    (void)in_sizes; (void)n_in; (void)out_size; (void)ws_size;
    const float* h          = (const float*)d_in[0];
    const float* att_feats1 = (const float*)d_in[1];
    const float* att_feats2 = (const float*)d_in[2];
    const float* att_masks  = (const float*)d_in[3];
    const float* W_h        = (const float*)d_in[4];
    const float* b_h        = (const float*)d_in[5];
    const float* w_a        = (const float*)d_in[6];
    const float* b_a        = (const float*)d_in[7];
    float* out   = (float*)d_out;
    float* att_h = (float*)d_ws;   // 256*512 floats = 512 KB scratch

    h2att_wmma_kernel<<<64, 256, 0, stream>>>(h, W_h, b_h, att_h);
    attn_fused_kernel<<<B_, 256, 0, stream>>>(att_feats1, att_feats2, att_masks,
                                              att_h, w_a, b_a, out);
}